// InputEmbedding_21741124452593
// MI455X (gfx1250) — compile-verified
//
#include <hip/hip_runtime.h>
#include <stdint.h>

typedef float v2f __attribute__((ext_vector_type(2)));
typedef float v8f __attribute__((ext_vector_type(8)));
typedef float f4  __attribute__((ext_vector_type(4)));

#define SEQ_L 1024
#define CM    256
#define CZ    128
#define NREL  65

// ---------------------------------------------------------------------------
// msa = one_hot(clip(seq)) @ W_msa + b_msa   -- exact via f32 WMMA (16x16x4),
// K = 21 padded to 24. Branch-free: B rows for k>=21 are loaded from a clamped
// (in-bounds) index and zeroed with a select, so no EXEC predication is ever
// generated around the WMMAs (which require EXEC == all ones).
// ---------------------------------------------------------------------------
__global__ __launch_bounds__(32) void msa_wmma_kernel(
    const int* __restrict__ seq,
    const float* __restrict__ W,     // [21, CM]
    const float* __restrict__ bias,  // [CM]
    float* __restrict__ out)         // [SEQ_L, CM]
{
  const int lane = threadIdx.x;      // 0..31
  const int half = lane >> 4;        // 0/1
  const int ml   = lane & 15;
  const int mt   = blockIdx.x;       // SEQ_L/16 = 64
  const int nt   = blockIdx.y;       // CM/16   = 16
  const int n    = nt * 16 + ml;

  int s = seq[mt * 16 + ml];
  s = s < 0 ? 0 : (s > 20 ? 20 : s);

  v8f acc = {0.f, 0.f, 0.f, 0.f, 0.f, 0.f, 0.f, 0.f};
#pragma unroll
  for (int k0 = 0; k0 < 24; k0 += 4) {
    const int ka  = k0 + 2 * half;           // A: M=ml, K = ka (vgpr0), ka+1 (vgpr1)
    const int ka0 = ka     < 21 ? ka     : 20;   // clamped, always in-bounds
    const int ka1 = ka + 1 < 21 ? ka + 1 : 20;
    const float w0 = W[ka0 * CM + n];        // unconditional loads
    const float w1 = W[ka1 * CM + n];
    v2f a, b;
    a.x = (s == ka    ) ? 1.0f : 0.0f;
    a.y = (s == ka + 1) ? 1.0f : 0.0f;
    b.x = (ka     < 21) ? w0 : 0.0f;         // v_cndmask, no branch
    b.y = (ka + 1 < 21) ? w1 : 0.0f;
    acc = __builtin_amdgcn_wmma_f32_16x16x4_f32(false, a, false, b,
                                                (short)0, acc, false, false);
  }
  const float bn = bias[n];
#pragma unroll
  for (int r = 0; r < 8; ++r) {
    const int m = mt * 16 + r + 8 * half;    // C/D layout: rows r / r+8
    out[m * CM + n] = acc[r] + bn;
  }
}

// ---------------------------------------------------------------------------
// pair[i,j,:] = (W_left[s_i]+b_left) + (W_right[s_j]+b_right) + relpos[d(i,j)]
// Pure store-bandwidth kernel: every operand staged into LDS, NT b128 stores.
// Block = 256 threads (8 waves), covers one i and 128 j's (64 KB of output).
// relpos table staged with CDNA5 async global->LDS loads (ASYNCcnt).
// ---------------------------------------------------------------------------
__global__ __launch_bounds__(256) void pair_kernel(
    const int* __restrict__ seq, const int* __restrict__ ri,
    const float* __restrict__ W_left,  const float* __restrict__ b_left,
    const float* __restrict__ W_right, const float* __restrict__ b_right,
    const float* __restrict__ relpos,  // [65, CZ]
    float* __restrict__ pair_out)      // [SEQ_L, SEQ_L, CZ]
{
  __shared__ __align__(16) float relpos_lds[NREL * CZ];  // 33280 B
  __shared__ __align__(16) float rtab_lds[21 * CZ];      // 10752 B (21 distinct right rows)
  __shared__ int sj_lds[128];
  __shared__ int d_lds[128];

  const int t    = threadIdx.x;
  const int i    = blockIdx.y;
  const int j0   = blockIdx.x * 128;
  const int c4   = (t & 31) * 4;   // this lane's channel quad
  const int wsub = t >> 5;         // wave id 0..7 (wave-uniform)

  // (1) Async-stage the whole relpos table into LDS (2080 x b128).
  {
    const uint32_t lbase = (uint32_t)(uintptr_t)&relpos_lds[0]; // low 32b of flat = LDS offset
    for (int idx = t; idx < NREL * CZ / 4; idx += 256) {
      const uint64_t ga = (uint64_t)(uintptr_t)(relpos + idx * 4);
      const uint32_t la = lbase + (uint32_t)idx * 16u;
      asm volatile("global_load_async_to_lds_b128 %0, %1, off"
                   :: "v"(la), "v"(ga) : "memory");
    }
  }

  // (2) Precompute the 21 distinct right rows (W_right[s] + b_right) into LDS.
  for (int idx = t; idx < 21 * CZ / 4; idx += 256) {
    const int c = (idx & 31) * 4;                 // 32 quads per row
    f4 w = *(const f4*)(W_right + idx * 4);
    f4 b = *(const f4*)(b_right + c);
    *(f4*)(rtab_lds + idx * 4) = w + b;
  }

  // (3) Per-j metadata: clipped sequence id and relpos bucket d.
  const int rii = ri[i];
  if (t < 128) {
    int sj = seq[j0 + t];
    sj_lds[t] = sj < 0 ? 0 : (sj > 20 ? 20 : sj);
    int d = rii - ri[j0 + t] + 32;
    d_lds[t] = d < 0 ? 0 : (d > 64 ? 64 : d);
  }

  // Left row for this i (held in registers, exact f32).
  int si = seq[i]; si = si < 0 ? 0 : (si > 20 ? 20 : si);
  f4 l4 = *(const f4*)(W_left + si * CZ + c4) + *(const f4*)(b_left + c4);

  asm volatile("s_wait_asynccnt 0" ::: "memory");  // my wave's async fills done
  __syncthreads();                                 // everyone's fills visible

  // (4) Stream 64 KB of output: each wave owns one j per iteration,
  //     32 lanes x float4 = 512 B contiguous, non-temporal (bypass L2 dirty).
  const size_t base = ((size_t)i * SEQ_L + j0) * CZ;
#pragma unroll 4
  for (int jj = 0; jj < 16; ++jj) {
    const int jl = jj * 8 + wsub;
    // jl (and thus sj/d) is wave-uniform: pin to SGPRs so row bases fold into
    // scalar address math instead of per-lane VALU chains.
    const int sj = __builtin_amdgcn_readfirstlane(sj_lds[jl]);
    const int d  = __builtin_amdgcn_readfirstlane(d_lds[jl]);
    const f4 r4 = *(const f4*)(rtab_lds   + sj * CZ + c4);
    const f4 p4 = *(const f4*)(relpos_lds + d  * CZ + c4);
    f4 o = l4 + r4 + p4;
    __builtin_nontemporal_store(o, (f4*)(pair_out + base + (size_t)jl * CZ + c4));
  }
}

// ---------------------------------------------------------------------------
extern "C" void kernel_launch(void* const* d_in, const int* in_sizes, int n_in,
                              void* d_out, int out_size, void* d_ws, size_t ws_size,
                              hipStream_t stream) {
  const int*   seq     = (const int*)  d_in[0];
  const int*   ri      = (const int*)  d_in[1];
  const float* W_msa   = (const float*)d_in[2];
  const float* b_msa   = (const float*)d_in[3];
  const float* W_left  = (const float*)d_in[4];
  const float* b_left  = (const float*)d_in[5];
  const float* W_right = (const float*)d_in[6];
  const float* b_right = (const float*)d_in[7];
  const float* relpos  = (const float*)d_in[8];

  float* out  = (float*)d_out;
  float* msa  = out;                        // [1, 1024, 256]
  float* pair = out + (size_t)SEQ_L * CM;   // [1024, 1024, 128]

  msa_wmma_kernel<<<dim3(SEQ_L / 16, CM / 16), 32, 0, stream>>>(seq, W_msa, b_msa, msa);
  pair_kernel<<<dim3(SEQ_L / 128, SEQ_L), 256, 0, stream>>>(
      seq, ri, W_left, b_left, W_right, b_right, relpos, pair);
}